// AdaptiveRegionalEdgeDiceCLDiceLoss_16647293240107
// MI455X (gfx1250) — compile-verified
//
#include <hip/hip_runtime.h>
#include <math.h>

// AdaptiveRegionalEdgeDiceCLDiceLoss for MI455X (gfx1250).
// pred/gt: (2,1,192,192,192) f32. One 256-thread WG per 16^3 block (3456 blocks).
// Thread (z,x) keeps its y-row of 16 floats in registers; x/z stencil exchange via
// LDS b128 rows; blocks loaded with CDNA5 async global->LDS DMA (ASYNCcnt path).

#define DIMN   192
#define SLICE  (DIMN*DIMN)          // 36864
#define VOL    (DIMN*DIMN*DIMN)     // 7077888
#define NBLK   3456                 // 2 * 12^3
#define ROWS   20                   // padded LDS row stride (floats), 16B-aligned rows
#define FSZ    (256*ROWS)           // floats per LDS field buffer (20KB)
#define ADPS   1e-8f

// ---------------- CDNA5 async global->LDS path ----------------
#if __has_builtin(__builtin_amdgcn_global_load_async_to_lds_b128)
#define ASYNC_BUILTIN 1
#else
#define ASYNC_BUILTIN 0
#endif
#if __has_builtin(__builtin_amdgcn_s_wait_asynccnt)
#define WAITA_BUILTIN 1
#else
#define WAITA_BUILTIN 0
#endif

typedef int v4i __attribute__((vector_size(4 * sizeof(int))));
typedef __attribute__((address_space(1))) v4i GV4;   // global int4
typedef __attribute__((address_space(3))) v4i LV4;   // LDS int4
typedef __attribute__((address_space(3))) void LAS;

__device__ __forceinline__ void asyncRow16(const float* g, float* l) {
#if ASYNC_BUILTIN
#pragma unroll
  for (int i = 0; i < 4; ++i)
    __builtin_amdgcn_global_load_async_to_lds_b128((GV4*)(g + 4*i), (LV4*)(l + 4*i), 0, 0);
#else
#pragma unroll
  for (int i = 0; i < 4; ++i) {
    unsigned lo = (unsigned)(size_t)(LAS*)(l + 4*i);
    asm volatile("global_load_async_to_lds_b128 %0, %1, off"
                 :: "v"(lo), "v"(g + 4*i) : "memory");
  }
#endif
}

__device__ __forceinline__ void waitAsync() {
#if WAITA_BUILTIN
  __builtin_amdgcn_s_wait_asynccnt(0);
#else
  asm volatile("s_wait_asynccnt 0" ::: "memory");
#endif
}

// ---------------- register-row helpers ----------------
struct Row { float v[16]; };

__device__ __forceinline__ void ldRow(const float* b, int zx, Row& r) {
  const float4* p = reinterpret_cast<const float4*>(b + zx*ROWS);
#pragma unroll
  for (int i = 0; i < 4; ++i) {
    float4 t = p[i];
    r.v[4*i+0] = t.x; r.v[4*i+1] = t.y; r.v[4*i+2] = t.z; r.v[4*i+3] = t.w;
  }
}
__device__ __forceinline__ void stRow(float* b, int zx, const Row& r) {
  float4* p = reinterpret_cast<float4*>(b + zx*ROWS);
#pragma unroll
  for (int i = 0; i < 4; ++i) {
    float4 t;
    t.x = r.v[4*i+0]; t.y = r.v[4*i+1]; t.z = r.v[4*i+2]; t.w = r.v[4*i+3];
    p[i] = t;
  }
}

// OP: 0=min(+inf pad -> skip), 1=max(-inf pad -> skip), 2=add(zero pad -> skip)
template<int OP> __device__ __forceinline__ float cmb(float a, float b) {
  if (OP == 0) return fminf(a, b);
  else if (OP == 1) return fmaxf(a, b);
  else return a + b;
}

// window-3 along y entirely in registers (min/max chains -> v_min3/v_max3)
template<int OP> __device__ __forceinline__ Row ywin(const Row& r) {
  Row o;
#pragma unroll
  for (int y = 0; y < 16; ++y) {
    float a = r.v[y];
    if (y > 0)  a = cmb<OP>(a, r.v[y-1]);
    if (y < 15) a = cmb<OP>(a, r.v[y+1]);
    o.v[y] = a;
  }
  return o;
}

template<int OP> __device__ __forceinline__ void nbX(const float* S, int z, int x, Row& a) {
  if (x > 0)  { Row t; ldRow(S, z*16 + x - 1, t);
#pragma unroll
    for (int y = 0; y < 16; ++y) a.v[y] = cmb<OP>(a.v[y], t.v[y]); }
  if (x < 15) { Row t; ldRow(S, z*16 + x + 1, t);
#pragma unroll
    for (int y = 0; y < 16; ++y) a.v[y] = cmb<OP>(a.v[y], t.v[y]); }
}
template<int OP> __device__ __forceinline__ void nbZ(const float* S, int z, int x, Row& a) {
  if (z > 0)  { Row t; ldRow(S, (z-1)*16 + x, t);
#pragma unroll
    for (int y = 0; y < 16; ++y) a.v[y] = cmb<OP>(a.v[y], t.v[y]); }
  if (z < 15) { Row t; ldRow(S, (z+1)*16 + x, t);
#pragma unroll
    for (int y = 0; y < 16; ++y) a.v[y] = cmb<OP>(a.v[y], t.v[y]); }
}

// soft_erode = 7-point min stencil (min of the three axis-wise 3-window minpools)
__device__ __forceinline__ Row erode7(const float* S, int z, int x, const Row& own) {
  Row a = ywin<0>(own);
  nbX<0>(S, z, x, a);
  nbZ<0>(S, z, x, a);
  return a;
}

// soft_dilate = separable 3x3x3 max (clamped edges). Field's own row given in regs.
__device__ __forceinline__ Row dilate27(const Row& e, float* T1, float* T2,
                                        int z, int x, int zx) {
  Row ty = ywin<1>(e);
  stRow(T1, zx, ty);
  __syncthreads();
  Row tx = ty; nbX<1>(T1, z, x, tx);
  stRow(T2, zx, tx);
  __syncthreads();
  Row d = tx; nbZ<1>(T2, z, x, d);
  return d;
}

// Laplacian boundary: b = 27*c - sum_3x3x3(zero pad); bit = (b > 0.1)
__device__ __forceinline__ unsigned lapBdr(const Row& c, float* T1, float* T2,
                                           int z, int x, int zx) {
  Row sy = ywin<2>(c);
  stRow(T1, zx, sy);
  __syncthreads();
  Row sx = sy; nbX<2>(T1, z, x, sx);
  stRow(T2, zx, sx);
  __syncthreads();
  Row s = sx; nbZ<2>(T2, z, x, s);
  unsigned bits = 0u;
#pragma unroll
  for (int y = 0; y < 16; ++y) {
    float b = fmaf(27.0f, c.v[y], -s.v[y]);
    bits |= (b > 0.1f) ? (1u << y) : 0u;
  }
  return bits;
}

// soft_skel with ITERS=3: needs E^1..E^4 and dilate(E^1..E^4).
// Loop intentionally NOT unrolled: keeps one erode/dilate body and short live ranges.
__device__ __forceinline__ void softSkel(const float* RAW, const Row& raw,
                                         float* A, float* Bb, float* Cc,
                                         int z, int x, int zx, Row& skel) {
  Row e = erode7(RAW, z, x, raw);       // E^1
  __syncthreads();
  stRow(A, zx, e);
  __syncthreads();
  Row img = raw;
  float* S  = A;                        // field holding E^(j+1)
  float* T1 = Bb;
  float* T2 = Cc;
#pragma unroll 1
  for (int j = 0; j < 4; ++j) {
    Row d = dilate27(e, T1, T2, z, x, zx);   // dilate(E^(j+1))
    if (j == 0) {
#pragma unroll
      for (int y = 0; y < 16; ++y) skel.v[y] = fmaxf(img.v[y] - d.v[y], 0.0f);
    } else {
#pragma unroll
      for (int y = 0; y < 16; ++y) {
        float dl = fmaxf(img.v[y] - d.v[y], 0.0f);
        skel.v[y] += fmaxf(dl - skel.v[y]*dl, 0.0f);
      }
    }
    img = e;
    if (j < 3) {
      Row e2 = erode7(S, z, x, e);      // E^(j+2)
      stRow(T1, zx, e2);                // T1's readers finished before T2-store barrier
      __syncthreads();
      float* ns = T1; T1 = S; S = ns;   // rotate buffers
      e = e2;
    }
  }
}

__device__ __forceinline__ float waveRed(float v) {
#pragma unroll
  for (int off = 16; off > 0; off >>= 1) v += __shfl_xor(v, off, 32);
  return v;
}

// ---------------- kernel 1: one workgroup per 16^3 block ----------------
// min 4 waves/SIMD => <=256 VGPRs/wave: no VGPR-MSB switching, 2 WGs/WGP resident.
__global__ __launch_bounds__(256, 4)
void arecl_block_kernel(const float* __restrict__ pred, const float* __restrict__ gt,
                        float* __restrict__ ws) {
  __shared__ float smem[4 * FSZ];       // 80 KB: rawP, rawG, 2 scratch
  __shared__ float red[8 * 9];
  float* B0 = smem;
  float* B1 = smem + FSZ;
  float* S0 = smem + 2*FSZ;
  float* S1 = smem + 3*FSZ;

  const int tid = threadIdx.x;
  const int z = tid >> 4, x = tid & 15, zx = tid;

  int m = blockIdx.x;
  int by = m % 12; int t = m / 12;
  int bx = t % 12; t /= 12;
  int bz = t % 12; int n = t / 12;
  size_t base = (size_t)n * VOL + (size_t)(bz*16 + z) * SLICE
              + (size_t)(bx*16 + x) * DIMN + (size_t)(by*16);

  // async DMA: global rows -> LDS (bypasses VGPRs, tracked by ASYNCcnt)
  asyncRow16(pred + base, B0 + zx*ROWS);
  asyncRow16(gt   + base, B1 + zx*ROWS);
  waitAsync();
  __syncthreads();

  // global dice partials (blocks tile the tensor exactly); pR not kept live
  float spg = 0.f, sp2 = 0.f, sg2 = 0.f;
  Row gR;
  {
    Row pR;
    ldRow(B0, zx, pR);
    ldRow(B1, zx, gR);
#pragma unroll
    for (int y = 0; y < 16; ++y) {
      spg = fmaf(pR.v[y], gR.v[y], spg);
      sp2 = fmaf(pR.v[y], pR.v[y], sp2);
      sg2 = fmaf(gR.v[y], gR.v[y], sg2);
    }
  }

  // ---- groundtruth phase ----
  unsigned gb = lapBdr(gR, S0, S1, z, x, zx);
  Row gcl;
  softSkel(B1, gR, S0, S1, B1, z, x, zx, gcl);   // raw-G buffer recycled as scratch
  __syncthreads();

  // ---- prediction phase (B0 untouched so far: reload pR) ----
  Row pR;
  ldRow(B0, zx, pR);
  unsigned pb = lapBdr(pR, S0, S1, z, x, zx);
  Row pcl;
  softSkel(B0, pR, S0, S1, B1, z, x, zx, pcl);

  // ---- per-block reductions ----
  float tb  = (float)__popc(pb & gb);            // sum p*g (binary bdr maps)
  float sbp = (float)__popc(pb);
  float sbg = (float)__popc(gb);
  float tpc = 0.f, spc = 0.f, sgc = 0.f;
#pragma unroll
  for (int y = 0; y < 16; ++y) {
    tpc = fmaf(pcl.v[y], gcl.v[y], tpc);
    spc += pcl.v[y];
    sgc += gcl.v[y];
  }

  float vals[9] = { spg, sp2, sg2, tb, sbp, sbg, tpc, spc, sgc };
#pragma unroll
  for (int k = 0; k < 9; ++k) vals[k] = waveRed(vals[k]);
  int wid = tid >> 5, lane = tid & 31;
  if (lane == 0) {
#pragma unroll
    for (int k = 0; k < 9; ++k) red[wid*9 + k] = vals[k];
  }
  __syncthreads();
  if (tid == 0) {
    float s[9];
#pragma unroll
    for (int k = 0; k < 9; ++k) s[k] = 0.f;
    for (int w = 0; w < 8; ++w)
#pragma unroll
      for (int k = 0; k < 9; ++k) s[k] += red[w*9 + k];

    // adaptive Tversky per block: fn = Sg - Sgp, fp = Sp - Sgp
    float tp = s[3], fp = s[4] - s[3], fn = s[5] - s[3];
    float den = fp + fn + ADPS;
    float alpha = 0.5f + 0.5f * ((fp + ADPS) / den);
    float beta  = 0.5f + 0.5f * ((fn + ADPS) / den);
    float lb = 1.f - (tp + ADPS) / (tp + alpha*fp + beta*fn + ADPS);

    float tc = s[6], fpc = s[7] - s[6], fnc = s[8] - s[6];
    float denc = fpc + fnc + ADPS;
    float ac = 0.5f + 0.5f * ((fpc + ADPS) / denc);
    float bc = 0.5f + 0.5f * ((fnc + ADPS) / denc);
    float lc = 1.f - (tc + ADPS) / (tc + ac*fpc + bc*fnc + ADPS);

    float* w = ws + (size_t)blockIdx.x * 8;
    w[0] = s[0]; w[1] = s[1]; w[2] = s[2]; w[3] = lb; w[4] = lc;
  }
}

// ---------------- kernel 2: deterministic final reduction ----------------
__global__ __launch_bounds__(256)
void arecl_finalize(const float* __restrict__ ws, const float* __restrict__ w1,
                    const float* __restrict__ w2, float* __restrict__ out) {
  __shared__ float red[8 * 5];
  float s[5] = {0.f, 0.f, 0.f, 0.f, 0.f};
  for (int m = threadIdx.x; m < NBLK; m += 256) {
    const float* w = ws + (size_t)m * 8;
#pragma unroll
    for (int k = 0; k < 5; ++k) s[k] += w[k];
  }
#pragma unroll
  for (int k = 0; k < 5; ++k) s[k] = waveRed(s[k]);
  int wid = threadIdx.x >> 5, lane = threadIdx.x & 31;
  if (lane == 0) {
#pragma unroll
    for (int k = 0; k < 5; ++k) red[wid*5 + k] = s[k];
  }
  __syncthreads();
  if (threadIdx.x == 0) {
    float a[5] = {0.f, 0.f, 0.f, 0.f, 0.f};
    for (int w = 0; w < 8; ++w)
#pragma unroll
      for (int k = 0; k < 5; ++k) a[k] += red[w*5 + k];

    float dice = 2.0f * a[0] / fmaxf(a[1] + a[2], 1e-6f);
    float dice_loss = 1.0f - dice;
    float w1s = w1[0], w2s = w2[0];
    float edge = (a[3] / (w1s*w1s) + a[4] / (w2s*w2s)) / (2.0f * (float)NBLK)
               + logf(1.0f + fabsf(w1s) * fabsf(w2s));
    out[0] = (dice < 0.8f) ? dice_loss : (dice_loss + edge);
  }
}

extern "C" void kernel_launch(void* const* d_in, const int* in_sizes, int n_in,
                              void* d_out, int out_size, void* d_ws, size_t ws_size,
                              hipStream_t stream) {
  const float* pred = (const float*)d_in[0];
  const float* gt   = (const float*)d_in[1];
  const float* w1   = (const float*)d_in[2];
  const float* w2   = (const float*)d_in[3];
  float* ws = (float*)d_ws;            // needs NBLK*8*4 = 110,592 bytes
  (void)in_sizes; (void)n_in; (void)out_size; (void)ws_size;

  hipLaunchKernelGGL(arecl_block_kernel, dim3(NBLK), dim3(256), 0, stream, pred, gt, ws);
  hipLaunchKernelGGL(arecl_finalize, dim3(1), dim3(256), 0, stream, ws, w1, w2, (float*)d_out);
}